// ExtendedKalmanFilter_18923625906192
// MI455X (gfx1250) — compile-verified
//
#include <hip/hip_runtime.h>

// EKF scan: B=4096 batches, T=256 steps, m=8 state, n=4 obs.
// 2 batches per wave, packed block-diagonally into 16x16 f32 WMMA tiles.
// All matrix products use V_WMMA_F32_16X16X4_F32 (4 chained K=4 chunks).

#define Bn 4096
#define Tn 256
#define WPB 8              // waves per block
#define BLOCK (WPB * 32)

typedef __attribute__((ext_vector_type(2))) float v2f;
typedef __attribute__((ext_vector_type(8))) float v8f;

struct WaveBuf {
  float F[256];   // block-diag F (16x16, row-major)
  float H[256];   // block-diag H (rows 0-3 / 8-11 live)
  float P[256];   // P_post -> P_prior -> P_new staging
  float Tm[256];  // generic intermediate staging
  float S[256];   // innovation covariance
  float SI[256];  // S^-1 (block 4x4s; rest stays 0)
  float K[256];   // Kalman gain staging
  float xq[16];   // x_post  (2 batches x 8)
  float xp[16];   // x_prior
  float yp[16];   // y_pred  (2 batches x 4)
  float dy[16];   // innovation
};

// Intra-wave LDS producer->consumer ordering. DS ops from one wave complete
// in order (CDNA5 ISA 7.3), so we only need the hw wait + a compiler fence.
__device__ __forceinline__ void wfence() {
  asm volatile("s_wait_dscnt 0x0" ::: "memory");
  __builtin_amdgcn_wave_barrier();
}

// D = A(16x16 from Am, row-major LDS) * B(16x16 from Bm) + acc, via 4x K=4 WMMA.
// BT=true uses Bm transposed (reads columns of Bm as B rows -> contiguous).
template <bool BT>
__device__ __forceinline__ v8f wmma_mac(const float* __restrict__ Am,
                                        const float* __restrict__ Bm,
                                        v8f acc, int mn, int hh) {
#pragma unroll
  for (int kb = 0; kb < 16; kb += 4) {
    const int k0 = kb + hh * 2;
    v2f a, b;
    a[0] = Am[mn * 16 + k0];
    a[1] = Am[mn * 16 + k0 + 1];
    if (BT) {               // B = Bm^T : row k of B is column k of Bm
      b[0] = Bm[mn * 16 + k0];
      b[1] = Bm[mn * 16 + k0 + 1];
    } else {
      b[0] = Bm[k0 * 16 + mn];
      b[1] = Bm[(k0 + 1) * 16 + mn];
    }
    acc = __builtin_amdgcn_wmma_f32_16x16x4_f32(false, a, false, b,
                                                (short)0, acc, false, false);
  }
  return acc;
}

// Scatter a C/D-layout accumulator to row-major LDS (optionally negated).
__device__ __forceinline__ void stage(float* __restrict__ Mm, v8f c, int cc,
                                      int hh, float sgn = 1.0f) {
#pragma unroll
  for (int r = 0; r < 8; ++r) Mm[(r + hh * 8) * 16 + cc] = sgn * c[r];
}

__global__ __launch_bounds__(BLOCK) void ekf_wmma_kernel(
    const float* __restrict__ yin, const float* __restrict__ m1x0,
    const float* __restrict__ m2x0, const float* __restrict__ Ain,
    const float* __restrict__ Cin, const float* __restrict__ Qin,
    const float* __restrict__ Rin, float* __restrict__ out) {
  __shared__ WaveBuf wb[WPB];
  __shared__ float sA[64], sC[32], sQ[64], sR[16];

  const int tid = threadIdx.x;
  if (tid < 64) sA[tid] = Ain[tid];
  if (tid < 32) sC[tid] = Cin[tid];
  if (tid >= 64 && tid < 128) sQ[tid - 64] = Qin[tid - 64];
  if (tid >= 128 && tid < 144) sR[tid - 128] = Rin[tid - 128];
  __syncthreads();

  const int wv = tid >> 5;
  const int lane = tid & 31;
  const int hh = lane >> 4;        // half-wave: which 8 rows of C/D I hold
  const int cc = lane & 15;        // my column (B/C/D) == my row (A operand)
  const int b0 = (blockIdx.x * WPB + wv) * 2;  // first of my 2 batches
  WaveBuf& w = wb[wv];

  const int i8 = lane & 7;         // state index     (lanes 0..15 canonical)
  const int bs8 = (lane >> 3) & 1; // batch select for state lanes
  const int j4 = lane & 3;         // obs index       (lanes 0..7 canonical)
  const int bs4 = (lane >> 2) & 1; // batch select for obs lanes
  const bool inb = ((cc >> 3) == hh);  // my C/D entries are in-block

  // ---- init carries: x0, P0 (block-diag), S^-1 buffer zeroed ----
  if (lane < 16) w.xq[lane] = m1x0[i8];
  v8f qc, rc;
#pragma unroll
  for (int r = 0; r < 8; ++r) {
    const int row = r + hh * 8;
    const int li = row & 7, lj = cc & 7;
    qc[r] = inb ? sQ[li * 8 + lj] : 0.0f;
    rc[r] = (inb && li < 4 && lj < 4) ? sR[li * 4 + lj] : 0.0f;
    w.P[row * 16 + cc] = inb ? m2x0[li * 8 + lj] : 0.0f;
    w.SI[row * 16 + cc] = 0.0f;
  }
  wfence();

  const size_t S_OFF = (size_t)Bn * Tn * 8;
  const size_t K_OFF = S_OFF + (size_t)Bn * Tn * 64;
  const v8f vzero = {0.f, 0.f, 0.f, 0.f, 0.f, 0.f, 0.f, 0.f};

  for (int t = 0; t < Tn; ++t) {
    // observation (lanes 0..7 canonical; all lanes load duplicates)
    const float yobs = yin[((size_t)(b0 + bs4) * Tn + t) * 4 + j4];
    if (t + 1 < Tn)
      __builtin_prefetch(&yin[((size_t)(b0 + bs4) * Tn + (t + 1)) * 4 + j4], 0, 3);

    // ---- predict mean: x_prior = A x + 0.1 sin(x); F = A + 0.1 diag(cos x)
    const float xl = w.xq[cc];               // x_post[my diag index]
    float xpv = 0.1f * sinf(xl);
#pragma unroll
    for (int j = 0; j < 8; ++j) xpv += sA[i8 * 8 + j] * w.xq[bs8 * 8 + j];
    if (lane < 16) w.xp[lane] = xpv;
    const float df = 0.1f * cosf(xl);
#pragma unroll
    for (int r = 0; r < 8; ++r) {
      const int row = r + hh * 8;
      float v = 0.0f;
      if (inb) {
        v = sA[(row & 7) * 8 + (cc & 7)];
        if (row == cc) v += df;
      }
      w.F[row * 16 + cc] = v;
    }
    wfence();

    // ---- G = F * P_post ---- ; y_pred = tanh(C x_prior), dy
    v8f g = wmma_mac<false>(w.F, w.P, vzero, cc, hh);
    float yacc = 0.0f;
#pragma unroll
    for (int j = 0; j < 8; ++j) yacc += sC[j4 * 8 + j] * w.xp[bs4 * 8 + j];
    const float yv = tanhf(yacc);
    if (lane < 8) {
      w.yp[lane] = yv;
      w.dy[lane] = yobs - yv;
    }
    stage(w.Tm, g, cc, hh);
    wfence();

    // ---- P_prior = G * F^T + Q ---- ; H = diag(1 - y^2) C (block-diag)
    v8f pp = wmma_mac<true>(w.Tm, w.F, qc, cc, hh);
#pragma unroll
    for (int r = 0; r < 8; ++r) {
      const int row = r + hh * 8;
      float v = 0.0f;
      if (inb && (row & 7) < 4) {
        const float yr = w.yp[(row >> 3) * 4 + (row & 7)];
        v = (1.0f - yr * yr) * sC[(row & 7) * 8 + (cc & 7)];
      }
      w.H[row * 16 + cc] = v;
    }
    stage(w.P, pp, cc, hh);   // P buffer now holds P_prior
    wfence();

    // ---- T2 = H * P_prior ----
    v8f t2 = wmma_mac<false>(w.H, w.P, vzero, cc, hh);
    stage(w.Tm, t2, cc, hh);
    wfence();

    // ---- S = T2 * H^T + R ----
    v8f s = wmma_mac<true>(w.Tm, w.H, rc, cc, hh);
    stage(w.S, s, cc, hh);
    wfence();

    // ---- S^-1 (4x4 adjugate; 32 lanes = 2 batches x 16 entries) ----
    {
      const int bsel = lane >> 4;
      const int idx = lane & 15;
      const int io = idx >> 2, jo = idx & 3;
      const float* Sb = &w.S[bsel * 136];  // (bsel*8)*16 + bsel*8
#define AE(r, c) Sb[(r)*16 + (c)]
      const float a00 = AE(0, 0), a01 = AE(0, 1), a02 = AE(0, 2), a03 = AE(0, 3);
      const float a10 = AE(1, 0), a11 = AE(1, 1), a12 = AE(1, 2), a13 = AE(1, 3);
      const float a20 = AE(2, 0), a21 = AE(2, 1), a22 = AE(2, 2), a23 = AE(2, 3);
      const float a30 = AE(3, 0), a31 = AE(3, 1), a32 = AE(3, 2), a33 = AE(3, 3);
#undef AE
      const float s0 = a00 * a11 - a10 * a01, s1 = a00 * a12 - a10 * a02;
      const float s2 = a00 * a13 - a10 * a03, s3 = a01 * a12 - a11 * a02;
      const float s4 = a01 * a13 - a11 * a03, s5 = a02 * a13 - a12 * a03;
      const float c5 = a22 * a33 - a32 * a23, c4 = a21 * a33 - a31 * a23;
      const float c3 = a21 * a32 - a31 * a22, c2 = a20 * a33 - a30 * a23;
      const float c1 = a20 * a32 - a30 * a22, c0 = a20 * a31 - a30 * a21;
      const float rdet =
          1.0f / (s0 * c5 - s1 * c4 + s2 * c3 + s3 * c2 - s4 * c1 + s5 * c0);
      const float b00 = a11 * c5 - a12 * c4 + a13 * c3;
      const float b01 = -a01 * c5 + a02 * c4 - a03 * c3;
      const float b02 = a31 * s5 - a32 * s4 + a33 * s3;
      const float b03 = -a21 * s5 + a22 * s4 - a23 * s3;
      const float b10 = -a10 * c5 + a12 * c2 - a13 * c1;
      const float b11 = a00 * c5 - a02 * c2 + a03 * c1;
      const float b12 = -a30 * s5 + a32 * s2 - a33 * s1;
      const float b13 = a20 * s5 - a22 * s2 + a23 * s1;
      const float b20 = a10 * c4 - a11 * c2 + a13 * c0;
      const float b21 = -a00 * c4 + a01 * c2 - a03 * c0;
      const float b22 = a30 * s4 - a31 * s2 + a33 * s0;
      const float b23 = -a20 * s4 + a21 * s2 - a23 * s0;
      const float b30 = -a10 * c3 + a11 * c1 - a12 * c0;
      const float b31 = a00 * c3 - a01 * c1 + a02 * c0;
      const float b32 = -a30 * s3 + a31 * s1 - a32 * s0;
      const float b33 = a20 * s3 - a21 * s1 + a22 * s0;
      const float r0v = (jo == 0) ? b00 : (jo == 1) ? b01 : (jo == 2) ? b02 : b03;
      const float r1v = (jo == 0) ? b10 : (jo == 1) ? b11 : (jo == 2) ? b12 : b13;
      const float r2v = (jo == 0) ? b20 : (jo == 1) ? b21 : (jo == 2) ? b22 : b23;
      const float r3v = (jo == 0) ? b30 : (jo == 1) ? b31 : (jo == 2) ? b32 : b33;
      const float iv =
          ((io == 0) ? r0v : (io == 1) ? r1v : (io == 2) ? r2v : r3v) * rdet;
      w.SI[(bsel * 8 + io) * 16 + (bsel * 8 + jo)] = iv;
    }
    // ---- T3 = P_prior * H^T ----
    v8f t3 = wmma_mac<true>(w.P, w.H, vzero, cc, hh);
    stage(w.Tm, t3, cc, hh);
    wfence();

    // ---- K = T3 * S^-1 ----
    v8f kk = wmma_mac<false>(w.Tm, w.SI, vzero, cc, hh);
    stage(w.K, kk, cc, hh);
    wfence();

    // ---- x_new = x_prior + K dy ----
    float xnv = xpv;
#pragma unroll
    for (int j = 0; j < 4; ++j)
      xnv += w.K[(bs8 * 8 + i8) * 16 + (bs8 * 8 + j)] * w.dy[bs8 * 4 + j];
    if (lane < 16) {
      w.xq[lane] = xnv;
      __builtin_nontemporal_store(
          xnv, &out[((size_t)(b0 + bs8) * Tn + t) * 8 + i8]);
    }

    // ---- T4 = K * S (store negated) ----
    v8f t4 = wmma_mac<false>(w.K, w.S, vzero, cc, hh);
    stage(w.Tm, t4, cc, hh, -1.0f);
    wfence();

    // ---- P_new = (-T4) * K^T + P_prior ----
    v8f pn = wmma_mac<true>(w.Tm, w.K, pp, cc, hh);
    stage(w.P, pn, cc, hh);   // carry for next step

    // ---- stream sigma and K outputs (non-temporal; 436MB > L2) ----
    if (inb) {
      const int lc = cc & 7;
      float* sp = out + S_OFF + ((size_t)(b0 + hh) * Tn + t) * 64;
#pragma unroll
      for (int r = 0; r < 8; ++r)
        __builtin_nontemporal_store(pn[r], &sp[r * 8 + lc]);
      if (lc < 4) {
        float* kp = out + K_OFF + ((size_t)(b0 + hh) * Tn + t) * 32;
#pragma unroll
        for (int r = 0; r < 8; ++r)
          __builtin_nontemporal_store(kk[r], &kp[r * 4 + lc]);
      }
    }
    wfence();  // xq / P visible for next iteration
  }
}

extern "C" void kernel_launch(void* const* d_in, const int* in_sizes, int n_in,
                              void* d_out, int out_size, void* d_ws,
                              size_t ws_size, hipStream_t stream) {
  (void)in_sizes; (void)n_in; (void)d_ws; (void)ws_size; (void)out_size;
  const float* y    = (const float*)d_in[0];
  const float* m1x0 = (const float*)d_in[1];
  const float* m2x0 = (const float*)d_in[2];
  const float* A    = (const float*)d_in[3];
  const float* C    = (const float*)d_in[4];
  const float* Q    = (const float*)d_in[5];
  const float* R    = (const float*)d_in[6];
  float* out = (float*)d_out;

  dim3 grid(Bn / (2 * WPB));  // 256 blocks x 256 threads = 2048 waves, 2 batches/wave
  ekf_wmma_kernel<<<grid, BLOCK, 0, stream>>>(y, m1x0, m2x0, A, C, Q, R, out);
}